// HungarianMatcher_4896262717957
// MI455X (gfx1250) — compile-verified
//
#include <hip/hip_runtime.h>
#include <hip/hip_bf16.h>
#include <math.h>

// ---- problem constants (from reference setup_inputs) ----
#define B_    32
#define Q_    900
#define QP_   912          // 57 * 16, query dim padded to WMMA tile
#define C_    91
#define VST_  96           // padded class stride of focal table (>=92, 16B aligned)
#define T_    64
#define TT_   2048         // B*T : global target columns of cm
#define QT_   57           // query tiles of 16 (912/16)
#define NT_   128          // target tiles of 16 (2048/16)
#define ALPHA_ 0.25f

static const size_t CM_ELEMS = (size_t)B_ * Q_ * TT_;   // 58,982,400 floats

typedef __attribute__((ext_vector_type(2))) float v2f;
typedef __attribute__((ext_vector_type(8))) float v8f;

// ------------------------------------------------------------------
// Kernel 1: focal-loss class-cost table  val[b][q][c] = pos - neg
// Only B*Q*C = 2.6M transcendental evaluations (vs 59M if fused).
// Padded rows (q>=900) and cols (c>=91) are zero so WMMA pad is inert.
// ------------------------------------------------------------------
__global__ void val_table_kernel(const float* __restrict__ logits,
                                 float* __restrict__ val) {
    int idx = blockIdx.x * blockDim.x + threadIdx.x;
    const int total = B_ * QP_ * VST_;
    if (idx >= total) return;
    int c = idx % VST_;
    int q = (idx / VST_) % QP_;
    int b = idx / (VST_ * QP_);
    float out = 0.0f;
    if (q < Q_ && c < C_) {
        float x = logits[((size_t)(b * Q_ + q)) * C_ + c];
        float p = 1.0f / (1.0f + __expf(-x));
        float om = 1.0f - p;
        float pos = ALPHA_ * om * om * (-__logf(p + 1e-8f));
        float neg = (1.0f - ALPHA_) * p * p * (-__logf(om + 1e-8f));
        out = pos - neg;
    }
    val[idx] = out;
}

// ------------------------------------------------------------------
// Kernel 2: cm[b][q][j] = val[b][q][tgt_id[j]] + L1(center_q, center_j)
// The gather over class ids is done as a dense f32 WMMA GEMM against a
// one-hot B matrix built in registers: exact (multipliers are 0.0/1.0).
// One wave per 16x16 output tile; K = 92 in 23 steps of V_WMMA_F32_16X16X4_F32.
// A-layout (ISA 7.12.2, 32-bit A 16x4): lane M = lane&15,
//   vgpr0/1 hold K = {0,1} (lanes 0-15) or K = {2,3} (lanes 16-31).
// C/D: vgpr i holds row M = i + 8*(lane>=16), col N = lane&15.
// ------------------------------------------------------------------
__global__ void cm_wmma_kernel(const float* __restrict__ val,
                               const float* __restrict__ pred_cp,
                               const int*   __restrict__ tgt_ids,
                               const float* __restrict__ tgt_cp,
                               float* __restrict__ cm) {
    int wave = blockIdx.x * (blockDim.x >> 5) + (threadIdx.x >> 5);
    int lane = threadIdx.x & 31;
    int tt = wave % NT_;                 // target tile (fastest: A reuse in WGP$)
    int qt = (wave / NT_) % QT_;         // query tile
    int b  = wave / (NT_ * QT_);         // batch
    int half = lane >> 4;                // 0: K/M low half, 1: high half
    int l15  = lane & 15;
    int kb   = half * 2;                 // K offset of this half's vgpr pair

    const float* arow = val + ((size_t)(b * QP_ + qt * 16 + l15)) * VST_ + kb;
    int j   = tt * 16 + l15;             // global target column for B/C/D
    int tid = tgt_ids[j];                // class id of this column

    v8f acc = {};
    for (int kk = 0; kk < 23; ++kk) {    // K = 0..91 in steps of 4
        v2f a = *(const v2f*)(arow + kk * 4);     // val[m][4kk+kb], val[m][4kk+kb+1]
        int k0 = kk * 4 + kb;
        v2f bb;
        bb.x = (tid == k0)     ? 1.0f : 0.0f;     // one-hot row K=k0
        bb.y = (tid == k0 + 1) ? 1.0f : 0.0f;     // one-hot row K=k0+1
        acc = __builtin_amdgcn_wmma_f32_16x16x4_f32(
            false, a, false, bb, (short)0, acc, false, false);
    }

    float tcx = tgt_cp[2 * j];
    float tcy = tgt_cp[2 * j + 1];
    int q0 = qt * 16 + half * 8;
#pragma unroll
    for (int i = 0; i < 8; ++i) {
        int q = q0 + i;
        if (q < Q_) {                    // divergence only after all WMMAs
            size_t row = (size_t)(b * Q_ + q);
            float qx = pred_cp[row * 2];
            float qy = pred_cp[row * 2 + 1];
            cm[row * TT_ + j] = acc[i] + fabsf(qx - tcx) + fabsf(qy - tcy);
        }
    }
}

// ------------------------------------------------------------------
// Fallback (no workspace): direct fused gather version. Store-bound too.
// ------------------------------------------------------------------
__global__ void cm_direct_kernel(const float* __restrict__ logits,
                                 const float* __restrict__ pred_cp,
                                 const int*   __restrict__ tgt_ids,
                                 const float* __restrict__ tgt_cp,
                                 float* __restrict__ cm, size_t total) {
    size_t idx = (size_t)blockIdx.x * blockDim.x + threadIdx.x;
    if (idx >= total) return;
    int j = (int)(idx % TT_);
    size_t bq = idx / TT_;
    int c = tgt_ids[j];
    float x = logits[bq * C_ + c];
    float p = 1.0f / (1.0f + __expf(-x));
    float om = 1.0f - p;
    float pos = ALPHA_ * om * om * (-__logf(p + 1e-8f));
    float neg = (1.0f - ALPHA_) * p * p * (-__logf(om + 1e-8f));
    float qx = pred_cp[bq * 2], qy = pred_cp[bq * 2 + 1];
    cm[idx] = (pos - neg) + fabsf(qx - tgt_cp[2 * j]) + fabsf(qy - tgt_cp[2 * j + 1]);
}

// ------------------------------------------------------------------
// Kernel 3: Jonker-Volgenant Hungarian, one workgroup per batch.
// 64x900 cost block (230 KB) lives entirely in CDNA5's 320 KB LDS.
// Inner O(m) scans parallelized over 256 threads + LDS tree argmin.
// ------------------------------------------------------------------
#define HUNG_SMEM_WORDS (64*900 + 65 + 901 + 901 + 901 + 901 + 901 + 256 + 256 + 64 + 4)

__global__ void hungarian_kernel(const float* __restrict__ cm,
                                 float* __restrict__ qout,
                                 float* __restrict__ tout) {
    extern __shared__ char smem[];
    float* costL = (float*)smem;               // [64][900] = cost (target x query)
    float* u     = costL + 64 * 900;           // [65]
    float* v     = u + 65;                     // [901]
    float* minv  = v + 901;                    // [901]
    int*   p     = (int*)(minv + 901);         // [901]
    int*   way   = p + 901;                    // [901]
    int*   used  = way + 901;                  // [901]
    float* rval  = (float*)(used + 901);       // [256]
    int*   ridx  = (int*)(rval + 256);         // [256]
    int*   t2q   = ridx + 256;                 // [64]
    int*   sj0   = t2q + 64;                   // [1]

    const int tid = threadIdx.x;
    const int b   = blockIdx.x;

    // stage cost[i][jq] = cm[b*Q+jq][b*T+i] (lanes sweep i -> coalesced global)
    for (int idx = tid; idx < 64 * 900; idx += 256) {
        int i  = idx & 63;
        int jq = idx >> 6;
        costL[i * 900 + jq] = cm[((size_t)(b * Q_ + jq)) * TT_ + b * T_ + i];
    }
    for (int jj = tid; jj <= 900; jj += 256) { v[jj] = 0.0f; p[jj] = 0; }
    if (tid <= 64) u[tid] = 0.0f;
    __syncthreads();

    for (int i = 1; i <= 64; ++i) {
        if (tid == 0) { p[0] = i; *sj0 = 0; }
        for (int jj = tid; jj <= 900; jj += 256) { minv[jj] = INFINITY; used[jj] = 0; }
        __syncthreads();

        for (;;) {
            int j0 = *sj0;
            int i0 = p[j0];
            if (tid == 0) used[j0] = 1;
            __syncthreads();

            float ui0 = u[i0];
            float lmin = INFINITY; int lj = 0;
            for (int jj = 1 + tid; jj <= 900; jj += 256) {
                if (!used[jj]) {
                    float cur = costL[(i0 - 1) * 900 + (jj - 1)] - ui0 - v[jj];
                    if (cur < minv[jj]) { minv[jj] = cur; way[jj] = j0; }
                    float mv = minv[jj];
                    if (mv < lmin) { lmin = mv; lj = jj; }
                }
            }
            rval[tid] = lmin; ridx[tid] = lj;
            __syncthreads();
            for (int s = 128; s > 0; s >>= 1) {
                if (tid < s && rval[tid + s] < rval[tid]) {
                    rval[tid] = rval[tid + s]; ridx[tid] = ridx[tid + s];
                }
                __syncthreads();
            }
            int j1 = ridx[0];
            float delta = rval[0];

            for (int jj = tid; jj <= 900; jj += 256) {
                if (used[jj]) { u[p[jj]] += delta; v[jj] -= delta; }   // p[jj] distinct
                else if (jj >= 1) minv[jj] -= delta;
            }
            if (tid == 0) *sj0 = j1;
            __syncthreads();
            if (p[j1] == 0) break;          // uniform
        }

        if (tid == 0) {                      // augment along way[]
            int j0 = *sj0;
            while (j0 != 0) { int j1 = way[j0]; p[j0] = p[j1]; j0 = j1; }
        }
        __syncthreads();
    }

    for (int jj = 1 + tid; jj <= 900; jj += 256) {
        int pi = p[jj];
        if (pi > 0) t2q[pi - 1] = jj - 1;
    }
    __syncthreads();
    if (tid < 64) {                          // rank = argsort position (values distinct)
        int myq = t2q[tid];
        int rank = 0;
        for (int s = 0; s < 64; ++s) rank += (t2q[s] < myq) ? 1 : 0;
        qout[b * T_ + rank] = (float)myq;
        tout[b * T_ + rank] = (float)tid;
    }
}

// ------------------------------------------------------------------
extern "C" void kernel_launch(void* const* d_in, const int* in_sizes, int n_in,
                              void* d_out, int out_size, void* d_ws, size_t ws_size,
                              hipStream_t stream) {
    const float* logits  = (const float*)d_in[0];   // [32,900,91] f32
    const float* pred_cp = (const float*)d_in[1];   // [32,900,2]  f32
    const int*   tgt_ids = (const int*)  d_in[2];   // [32,64]     int
    const float* tgt_cp  = (const float*)d_in[3];   // [32,64,2]   f32
    float* out = (float*)d_out;                     // cm | query_idx | target_idx

    const size_t need = (size_t)B_ * QP_ * VST_ * sizeof(float); // 10.7 MB focal table
    if (ws_size >= need) {
        float* val = (float*)d_ws;
        const int total1 = B_ * QP_ * VST_;
        val_table_kernel<<<(total1 + 255) / 256, 256, 0, stream>>>(logits, val);
        const int waves = B_ * QT_ * NT_;           // 233,472 tiles, 8 waves/block
        cm_wmma_kernel<<<waves / 8, 256, 0, stream>>>(val, pred_cp, tgt_ids, tgt_cp, out);
    } else {
        const size_t total = CM_ELEMS;
        cm_direct_kernel<<<(unsigned)((total + 255) / 256), 256, 0, stream>>>(
            logits, pred_cp, tgt_ids, tgt_cp, out, total);
    }

    const size_t smem = (size_t)HUNG_SMEM_WORDS * 4;   // ~251 KB < 320 KB LDS/WGP
    hungarian_kernel<<<B_, 256, smem, stream>>>(out, out + CM_ELEMS,
                                                out + CM_ELEMS + (size_t)B_ * T_);
}